// EdgeModel_19404662243985
// MI455X (gfx1250) — compile-verified
//
#include <hip/hip_runtime.h>
#include <math.h>

// EdgeModel: x = [src | dest | edge_attr | u[batch]]  (E x 384)
//            h = selu(x @ W1 + b1)                    (E x 32)
//            out = h @ W2 + b2                        (E x 64)
//
// f32 WMMA (16x16x4). Memory-bound (768 MB @ 23.3 TB/s => ~33us floor), so:
//  - keep f32 end-to-end (matches reference precision)
//  - weights pre-repacked into interleaved b128-per-lane fragment pairs
//  - each wave processes TWO 16-row M-tiles, reusing every weight fragment
//    (per K-chunk: 2 A-loads + 1 W-load + 4 WMMAs)
//  - streaming x loads / out stores are non-temporal so the 56 KB weight set
//    stays hot in WGP$/L2.

typedef __attribute__((ext_vector_type(2))) float v2f;
typedef __attribute__((ext_vector_type(4))) float v4f;
typedef __attribute__((ext_vector_type(8))) float v8f;

static constexpr int kNode = 128;
static constexpr int kEdge = 64;
static constexpr int kGlob = 64;
static constexpr int kHid  = 32;
static constexpr int kOut  = 64;

// Layer-1: 96 K-chunks, each a 128-float pair-block (32 lanes x 4 floats,
//          covering N-tiles 0 and 1).
static constexpr int kW1Blocks = 96;
// Layer-2: 8 K-chunks x 2 pair-blocks (cols 0-31 and 32-63).
static constexpr int kW2Blocks = 8 * 2;
static constexpr int kW2Base   = kW1Blocks * 128;   // float offset in ws

// ---------------------------------------------------------------------------
// Repack W1 [384,32] / W2 [32,64] (row-major, in x out) into interleaved
// WMMA B-fragment pair-blocks. Lane L (kk=L>>4, m=L&15), k0 = 4c + 2kk:
//   block[L*4 + 0..3] = { B[k0][colLo], B[k0+1][colLo], B[k0][colHi], B[k0+1][colHi] }
// ---------------------------------------------------------------------------
__global__ void repack_weights_kernel(const float* __restrict__ W1,
                                      const float* __restrict__ W2,
                                      float* __restrict__ wf) {
  int t = blockIdx.x * blockDim.x + threadIdx.x;
  const int totalW1 = kW1Blocks * 32;
  const int totalW2 = kW2Blocks * 32;
  if (t < totalW1) {
    int lane = t & 31;
    int c    = t >> 5;                 // 0..95
    int m    = lane & 15;
    int kk   = lane >> 4;
    int k0   = 4 * c + 2 * kk;
    float* o = wf + c * 128 + lane * 4;
    o[0] = W1[(size_t)k0 * kHid + m];              // N-tile 0
    o[1] = W1[(size_t)(k0 + 1) * kHid + m];
    o[2] = W1[(size_t)k0 * kHid + 16 + m];         // N-tile 1
    o[3] = W1[(size_t)(k0 + 1) * kHid + 16 + m];
  } else if (t < totalW1 + totalW2) {
    int t2   = t - totalW1;
    int lane = t2 & 31;
    int blk  = t2 >> 5;                // 0..15
    int c    = blk >> 1;               // K-chunk 0..7
    int half = blk & 1;                // column half: 0 -> cols 0-31, 1 -> 32-63
    int m    = lane & 15;
    int kk   = lane >> 4;
    int k0   = 4 * c + 2 * kk;
    int colLo = 32 * half + m;
    int colHi = 32 * half + 16 + m;
    float* o = wf + kW2Base + blk * 128 + lane * 4;
    o[0] = W2[(size_t)k0 * kOut + colLo];
    o[1] = W2[(size_t)(k0 + 1) * kOut + colLo];
    o[2] = W2[(size_t)k0 * kOut + colHi];
    o[3] = W2[(size_t)(k0 + 1) * kOut + colHi];
  }
}

// ---------------------------------------------------------------------------
// Main kernel: one wave per 32-edge pair-tile (two 16-row WMMA M-tiles),
// 8 waves per block.
// ---------------------------------------------------------------------------
__global__ __launch_bounds__(256)
void edge_mlp_kernel(const float* __restrict__ src,
                     const float* __restrict__ dst,
                     const float* __restrict__ ea,
                     const float* __restrict__ u,
                     const int*   __restrict__ batch,
                     const float* __restrict__ b1,
                     const float* __restrict__ b2,
                     const float* __restrict__ wf,
                     float* __restrict__ out,
                     int E, int numPairs) {
  // Per-wave private h staging (two 16x32 tiles) for C->A layout transpose.
  // Row stride 34 floats: ds_load_b64 stays 8B-aligned, near conflict-free.
  __shared__ float hbuf[16][16 * 34];

  const int lane = threadIdx.x & 31;
  const int wave = threadIdx.x >> 5;
  const int pair = blockIdx.x * 8 + wave;
  if (pair >= numPairs) return;            // wave-uniform => EXEC stays all-1s

  const int m  = lane & 15;
  const int kk = lane >> 4;
  const int base = pair * 32;
  int rowA = base + m;
  int rowB = base + 16 + m;
  if (rowA >= E) rowA = E - 1;             // clamp loads; stores are guarded
  if (rowB >= E) rowB = E - 1;

  const float* srcA = src + (size_t)rowA * kNode + 2 * kk;
  const float* srcB = src + (size_t)rowB * kNode + 2 * kk;
  const float* dstA = dst + (size_t)rowA * kNode + 2 * kk;
  const float* dstB = dst + (size_t)rowB * kNode + 2 * kk;
  const float* eaA  = ea  + (size_t)rowA * kEdge + 2 * kk;
  const float* eaB  = ea  + (size_t)rowB * kEdge + 2 * kk;
  const float* uA   = u + (size_t)batch[rowA] * kGlob + 2 * kk;
  const float* uB   = u + (size_t)batch[rowB] * kGlob + 2 * kk;
  const float* wp1  = wf + lane * 4;            // layer-1 pair-blocks
  const float* wp2  = wf + kW2Base + lane * 4;  // layer-2 pair-blocks

  // ---- GEMM1: two [16,384] tiles vs shared [384,32], bias-seeded ----------
  const float bia0 = b1[m];
  const float bia1 = b1[16 + m];
  v8f aA0 = {bia0, bia0, bia0, bia0, bia0, bia0, bia0, bia0};
  v8f aA1 = {bia1, bia1, bia1, bia1, bia1, bia1, bia1, bia1};
  v8f aB0 = aA0;
  v8f aB1 = aA1;

#define L1_STEP(PA, PB, CG, CL)                                                \
  do {                                                                         \
    v2f xa = __builtin_nontemporal_load((const v2f*)((PA) + 4 * (CL)));        \
    v2f xb = __builtin_nontemporal_load((const v2f*)((PB) + 4 * (CL)));        \
    v4f wq = *(const v4f*)(wp1 + (CG) * 128);                                  \
    v2f w0 = __builtin_shufflevector(wq, wq, 0, 1);                            \
    v2f w1 = __builtin_shufflevector(wq, wq, 2, 3);                            \
    aA0 = __builtin_amdgcn_wmma_f32_16x16x4_f32(false, xa, false, w0,          \
                                                (short)0, aA0, false, false);  \
    aA1 = __builtin_amdgcn_wmma_f32_16x16x4_f32(false, xa, false, w1,          \
                                                (short)0, aA1, false, false);  \
    aB0 = __builtin_amdgcn_wmma_f32_16x16x4_f32(false, xb, false, w0,          \
                                                (short)0, aB0, false, false);  \
    aB1 = __builtin_amdgcn_wmma_f32_16x16x4_f32(false, xb, false, w1,          \
                                                (short)0, aB1, false, false);  \
  } while (0)

#pragma unroll
  for (int c = 0; c < 32; ++c) L1_STEP(srcA, srcB, c, c);        // K 0..127
#pragma unroll
  for (int c = 0; c < 32; ++c) L1_STEP(dstA, dstB, 32 + c, c);   // K 128..255
#pragma unroll
  for (int c = 0; c < 16; ++c) L1_STEP(eaA, eaB, 64 + c, c);     // K 256..319
#pragma unroll
  for (int c = 0; c < 16; ++c) L1_STEP(uA, uB, 80 + c, c);       // K 320..383
#undef L1_STEP

  // ---- SELU ---------------------------------------------------------------
  const float kScale      = 1.0507009873554805f;
  const float kAlphaScale = 1.7580993408473766f;  // scale * alpha
#pragma unroll
  for (int i = 0; i < 8; ++i) {
    float v;
    v = aA0[i]; aA0[i] = v > 0.0f ? kScale * v : kAlphaScale * (expf(v) - 1.0f);
    v = aA1[i]; aA1[i] = v > 0.0f ? kScale * v : kAlphaScale * (expf(v) - 1.0f);
    v = aB0[i]; aB0[i] = v > 0.0f ? kScale * v : kAlphaScale * (expf(v) - 1.0f);
    v = aB1[i]; aB1[i] = v > 0.0f ? kScale * v : kAlphaScale * (expf(v) - 1.0f);
  }

  // ---- Transpose h (C-layout -> A-layout) through private LDS -------------
  float* hbA = hbuf[wave * 2];
  float* hbB = hbuf[wave * 2 + 1];
#pragma unroll
  for (int i = 0; i < 8; ++i) {
    int r = i + 8 * kk;                 // C-layout: VGPR i holds row i + 8*kk
    hbA[r * 34 + m]      = aA0[i];
    hbA[r * 34 + 16 + m] = aA1[i];
    hbB[r * 34 + m]      = aB0[i];
    hbB[r * 34 + 16 + m] = aB1[i];
  }

  // ---- GEMM2: two [16,32] tiles vs shared [32,64], bias-seeded ------------
  const float c0 = b2[m], c1 = b2[16 + m], c2 = b2[32 + m], c3 = b2[48 + m];
  v8f oA0 = {c0, c0, c0, c0, c0, c0, c0, c0};
  v8f oA1 = {c1, c1, c1, c1, c1, c1, c1, c1};
  v8f oA2 = {c2, c2, c2, c2, c2, c2, c2, c2};
  v8f oA3 = {c3, c3, c3, c3, c3, c3, c3, c3};
  v8f oB0 = oA0, oB1 = oA1, oB2 = oA2, oB3 = oA3;

  const float* hrA = hbA + m * 34 + 2 * kk;
  const float* hrB = hbB + m * 34 + 2 * kk;
#pragma unroll
  for (int c = 0; c < 8; ++c) {
    v2f xa = *(const v2f*)(hrA + 4 * c);
    v2f xb = *(const v2f*)(hrB + 4 * c);
    v4f qL = *(const v4f*)(wp2 + (c * 2 + 0) * 128);   // cols 0-31
    v4f qH = *(const v4f*)(wp2 + (c * 2 + 1) * 128);   // cols 32-63
    v2f w0 = __builtin_shufflevector(qL, qL, 0, 1);
    v2f w1 = __builtin_shufflevector(qL, qL, 2, 3);
    v2f w2 = __builtin_shufflevector(qH, qH, 0, 1);
    v2f w3 = __builtin_shufflevector(qH, qH, 2, 3);
    oA0 = __builtin_amdgcn_wmma_f32_16x16x4_f32(false, xa, false, w0, (short)0, oA0, false, false);
    oA1 = __builtin_amdgcn_wmma_f32_16x16x4_f32(false, xa, false, w1, (short)0, oA1, false, false);
    oA2 = __builtin_amdgcn_wmma_f32_16x16x4_f32(false, xa, false, w2, (short)0, oA2, false, false);
    oA3 = __builtin_amdgcn_wmma_f32_16x16x4_f32(false, xa, false, w3, (short)0, oA3, false, false);
    oB0 = __builtin_amdgcn_wmma_f32_16x16x4_f32(false, xb, false, w0, (short)0, oB0, false, false);
    oB1 = __builtin_amdgcn_wmma_f32_16x16x4_f32(false, xb, false, w1, (short)0, oB1, false, false);
    oB2 = __builtin_amdgcn_wmma_f32_16x16x4_f32(false, xb, false, w2, (short)0, oB2, false, false);
    oB3 = __builtin_amdgcn_wmma_f32_16x16x4_f32(false, xb, false, w3, (short)0, oB3, false, false);
  }

  // ---- Store out: VGPR i -> row (i + 8*kk), col (m + 16n), non-temporal ---
#pragma unroll
  for (int i = 0; i < 8; ++i) {
    int rA = base + i + 8 * kk;
    if (rA < E) {
      float* o = out + (size_t)rA * kOut + m;
      __builtin_nontemporal_store(oA0[i], o);
      __builtin_nontemporal_store(oA1[i], o + 16);
      __builtin_nontemporal_store(oA2[i], o + 32);
      __builtin_nontemporal_store(oA3[i], o + 48);
    }
    int rB = base + 16 + i + 8 * kk;
    if (rB < E) {
      float* o = out + (size_t)rB * kOut + m;
      __builtin_nontemporal_store(oB0[i], o);
      __builtin_nontemporal_store(oB1[i], o + 16);
      __builtin_nontemporal_store(oB2[i], o + 32);
      __builtin_nontemporal_store(oB3[i], o + 48);
    }
  }
}

extern "C" void kernel_launch(void* const* d_in, const int* in_sizes, int n_in,
                              void* d_out, int out_size, void* d_ws, size_t ws_size,
                              hipStream_t stream) {
  const float* src  = (const float*)d_in[0];
  const float* dst  = (const float*)d_in[1];
  const float* ea   = (const float*)d_in[2];
  const float* u    = (const float*)d_in[3];
  const int*   bat  = (const int*)d_in[4];
  const float* W1   = (const float*)d_in[5];
  const float* b1   = (const float*)d_in[6];
  const float* W2   = (const float*)d_in[7];
  const float* b2   = (const float*)d_in[8];
  float* out = (float*)d_out;
  float* wf  = (float*)d_ws;   // (96 + 16) * 128 floats = 57344 B

  const int E = in_sizes[0] / kNode;
  const int numPairs = (E + 31) / 32;

  {
    const int total = (kW1Blocks + kW2Blocks) * 32;
    repack_weights_kernel<<<(total + 255) / 256, 256, 0, stream>>>(W1, W2, wf);
  }
  {
    const int blocks = (numPairs + 7) / 8;   // 8 waves (pair-tiles) per block
    edge_mlp_kernel<<<blocks, 256, 0, stream>>>(src, dst, ea, u, bat,
                                                b1, b2, wf, out, E, numPairs);
  }
}